// feat_reg_ST_loss_10075993276815
// MI455X (gfx1250) — compile-verified
//
#include <hip/hip_runtime.h>
#include <cmath>

typedef float v2f __attribute__((ext_vector_type(2)));
typedef float v8f __attribute__((ext_vector_type(8)));

#if defined(__has_builtin)
#if __has_builtin(__builtin_amdgcn_wmma_f32_16x16x4_f32)
#define HAVE_WMMA_F32_K4 1
#endif
#endif

// Problem constants (from reference)
constexpr int BATCH = 2, FDIM = 2048, FH = 64, FW = 128, NCLS = 19, H = 512, W = 1024;
constexpr int HW = FH * FW;          // 8192 pixels per batch image
constexpr int NPIX = BATCH * HW;     // 16384 pixels per domain
constexpr int PW = 8;                // pixel-stripe parallelism per feature tile
constexpr int PIX_PER_STRIPE = NPIX / PW; // 2048
constexpr int LDST = 65;             // padded LDS row stride (bank-conflict free)

// Workspace layout (float units)
constexpr int OFF_SUM_S = 0;
constexpr int OFF_SUM_T = OFF_SUM_S + NCLS * FDIM;
constexpr int OFF_CENT  = OFF_SUM_T + NCLS * FDIM;
constexpr int OFF_SQ_S  = OFF_CENT  + NCLS * FDIM;
constexpr int OFF_SQ_T  = OFF_SQ_S + 32;
constexpr int OFF_CNT_S = OFF_SQ_T + 32;   // int
constexpr int OFF_CNT_T = OFF_CNT_S + 32;  // int
constexpr int OFF_SSQ_S = OFF_CNT_T + 32;
constexpr int OFF_SSQ_T = OFF_SSQ_S + 32;
constexpr int OFF_DIFFP = OFF_SSQ_T + 32;
constexpr int OFF_ZERO_END = OFF_DIFFP + 384;
constexpr int OFF_LAB_S = OFF_ZERO_END;    // int[NPIX]
constexpr int OFF_LAB_T = OFF_LAB_S + NPIX;

// ---------------------------------------------------------------- zero init
__global__ void k_zero(float* __restrict__ w, int n) {
    int i = blockIdx.x * blockDim.x + threadIdx.x;
    if (i < n) w[i] = 0.f;
}

// ------------------------------------------------- labels: bilinear + argmax
__global__ __launch_bounds__(256) void k_labels(const float* __restrict__ sm,
                                                int* __restrict__ labels,
                                                int* __restrict__ cnt) {
    int pid = blockIdx.x * blockDim.x + threadIdx.x;   // 0..NPIX-1
    if (pid >= NPIX) return;
    int b = pid >> 13;
    int rem = pid & (HW - 1);
    int y = rem >> 7, x = rem & 127;

    float fy = (float)y * ((float)(H - 1) / (float)(FH - 1));
    float fx = (float)x * ((float)(W - 1) / (float)(FW - 1));
    int y0 = (int)floorf(fy); if (y0 > H - 1) y0 = H - 1;
    int x0 = (int)floorf(fx); if (x0 > W - 1) x0 = W - 1;
    int y1 = min(y0 + 1, H - 1);
    int x1 = min(x0 + 1, W - 1);
    float wy = fy - (float)y0, wx = fx - (float)x0;

    float best = -3.402823e38f;
    int bi = 0;
    for (int c = 0; c < NCLS; ++c) {
        const float* p = sm + ((long)(b * NCLS + c)) * (long)(H * W);
        float v00 = p[y0 * W + x0], v01 = p[y0 * W + x1];
        float v10 = p[y1 * W + x0], v11 = p[y1 * W + x1];
        float v = ((1.f - wy) * v00 + wy * v10) * (1.f - wx) +
                  ((1.f - wy) * v01 + wy * v11) * wx;
        if (v > best) { best = v; bi = c; }
    }
    labels[pid] = bi;
    atomicAdd(&cnt[bi], 1);
}

// ------------------- single pass: one-hot WMMA segment-sum + per-class sum f^2
__global__ __launch_bounds__(256) void k_segsum(const float* __restrict__ feat,
                                                const int* __restrict__ labels,
                                                float* __restrict__ sums,
                                                float* __restrict__ sq) {
    __shared__ float tile[64 * LDST];
    __shared__ int   slab[64];
    __shared__ float sqbins[NCLS];

    const int tid = threadIdx.x;
    const int lane = tid & 31;
    const int wave = tid >> 5;
    const int f0 = blockIdx.x * 64;
    const int stripe = blockIdx.y;

    if (tid < NCLS) sqbins[tid] = 0.f;

    const int subtile   = wave & 3;        // which 16-feature column group
    const int classTile = wave >> 2;       // classes 0-15 / 16-31
    const int fs = subtile * 16;
    const int nlo = lane & 15;
    const int hi  = lane >> 4;             // 0: K={k,k+1}, 1: K={k+2,k+3}
    const int myClass = classTile * 16 + nlo;

    v8f acc = {0.f, 0.f, 0.f, 0.f, 0.f, 0.f, 0.f, 0.f};

    const int px  = tid & 63;
    const int fl0 = tid >> 6;              // 0..3

    for (int blk = 0; blk < PIX_PER_STRIPE / 64; ++blk) {
        const int gp0 = stripe * PIX_PER_STRIPE + blk * 64;
        if (tid < 64) slab[tid] = labels[gp0 + tid];
        __syncthreads();

        // stage 64px x 64feat tile (coalesced: lane = consecutive pixels)
        const int gp = gp0 + px;
        const int b  = gp >> 13;
        const long base = (long)b * ((long)FDIM * HW) + (gp & (HW - 1));
        float s2 = 0.f;
        #pragma unroll
        for (int i = 0; i < 16; ++i) {
            int fl = fl0 + i * 4;
            float v = feat[base + (long)(f0 + fl) * HW];
            tile[px * LDST + fl] = v;
            s2 += v * v;
        }
        atomicAdd(&sqbins[slab[px]], s2);
        __syncthreads();

        // D(class,feature) += onehot(class,pixel) x feat(pixel,feature), K=4 steps
        #pragma unroll
        for (int kb = 0; kb < 64; kb += 4) {
            const int k0 = kb + (hi << 1);
#ifdef HAVE_WMMA_F32_K4
            v2f a, bm;
            a.x  = (slab[k0]     == myClass) ? 1.f : 0.f;
            a.y  = (slab[k0 + 1] == myClass) ? 1.f : 0.f;
            bm.x = tile[k0 * LDST + fs + nlo];
            bm.y = tile[(k0 + 1) * LDST + fs + nlo];
            acc = __builtin_amdgcn_wmma_f32_16x16x4_f32(
                false, a, false, bm, (short)0, acc, false, false);
#else
            #pragma unroll
            for (int r = 0; r < 8; ++r) {
                const int cls = classTile * 16 + hi * 8 + r;
                #pragma unroll
                for (int k = 0; k < 4; ++k)
                    if (slab[kb + k] == cls)
                        acc[r] += tile[(kb + k) * LDST + fs + nlo];
            }
#endif
        }
        __syncthreads();
    }

    // flush D tile: lane holds D[classTile*16 + hi*8 + r][fs + nlo]
    const int mbase = classTile * 16 + hi * 8;
    const int fcol  = f0 + fs + nlo;
    #pragma unroll
    for (int r = 0; r < 8; ++r) {
        int cls = mbase + r;
        if (cls < NCLS) atomicAdd(&sums[cls * FDIM + fcol], acc[r]);
    }
    if (tid < NCLS) atomicAdd(&sq[tid], sqbins[tid]);
}

// ------------------- centroids + analytic per-class ssq (p==2 fast path)
__global__ __launch_bounds__(256) void k_centroid(const float* __restrict__ sum_s,
                                                  const float* __restrict__ sum_t,
                                                  const int* __restrict__ cnt_s,
                                                  const int* __restrict__ cnt_t,
                                                  const float* __restrict__ sq_s,
                                                  const float* __restrict__ sq_t,
                                                  float* __restrict__ cent,
                                                  float* __restrict__ ssq_s,
                                                  float* __restrict__ ssq_t,
                                                  const int* __restrict__ pnorm) {
    __shared__ float red[256];
    const int c = blockIdx.x;
    const int tid = threadIdx.x;
    const float den = fmaxf((float)(cnt_s[c] + cnt_t[c]), 1.f);
    float ds = 0.f, dt = 0.f, cn = 0.f;
    for (int f = tid; f < FDIM; f += 256) {
        float s = sum_s[c * FDIM + f];
        float t = sum_t[c * FDIM + f];
        float ce = (s + t) / den;
        cent[c * FDIM + f] = ce;
        ds += ce * s; dt += ce * t; cn += ce * ce;
    }
    float vals[3] = {ds, dt, cn}, out[3];
    for (int v = 0; v < 3; ++v) {
        red[tid] = vals[v]; __syncthreads();
        for (int s = 128; s > 0; s >>= 1) {
            if (tid < s) red[tid] += red[tid + s];
            __syncthreads();
        }
        out[v] = red[0]; __syncthreads();
    }
    if (tid == 0 && *pnorm == 2) {
        float ns = (float)cnt_s[c], nt = (float)cnt_t[c];
        ssq_s[c] = fmaxf(sq_s[c] - 2.f * out[0] + ns * out[2], 0.f);
        ssq_t[c] = fmaxf(sq_t[c] - 2.f * out[1] + nt * out[2], 0.f);
    }
}

// ------------------- pairwise centroid distances (19x19, any p)
__global__ __launch_bounds__(256) void k_pairwise(const float* __restrict__ cent,
                                                  float* __restrict__ diffp,
                                                  const int* __restrict__ pnorm) {
    __shared__ float red[256];
    const int i = blockIdx.x, j = blockIdx.y, tid = threadIdx.x;
    const int p = *pnorm;
    const float pf = (float)p;
    float a = 0.f;
    for (int f = tid; f < FDIM; f += 256) {
        float d = cent[i * FDIM + f] - cent[j * FDIM + f];
        a += (p == 2) ? d * d : powf(fabsf(d), pf);
    }
    red[tid] = a; __syncthreads();
    for (int s = 128; s > 0; s >>= 1) {
        if (tid < s) red[tid] += red[tid + s];
        __syncthreads();
    }
    if (tid == 0) diffp[i * NCLS + j] = red[0];
}

// ------------------- generic-p per-pixel pass (device no-op when p==2)
__global__ __launch_bounds__(256) void k_fdist_generic(const float* __restrict__ feat,
                                                       const int* __restrict__ labels,
                                                       const float* __restrict__ cent,
                                                       float* __restrict__ ssq,
                                                       const int* __restrict__ pnorm) {
    if (*pnorm == 2) return;   // analytic path already produced ssq
    const int pid = blockIdx.x * blockDim.x + threadIdx.x;
    if (pid >= NPIX) return;
    const int lab = labels[pid];
    const int b = pid >> 13;
    const float pf = (float)(*pnorm);
    const long base = (long)b * ((long)FDIM * HW) + (pid & (HW - 1));
    float dp = 0.f;
    for (int f = 0; f < FDIM; ++f) {
        float d = feat[base + (long)f * HW] - cent[lab * FDIM + f];
        dp += powf(fabsf(d), pf);
    }
    atomicAdd(&ssq[lab], dp);
}

// ------------------- finalize: 3 scalar outputs
__global__ void k_final(const int* __restrict__ cnt_s, const int* __restrict__ cnt_t,
                        const float* __restrict__ ssq_s, const float* __restrict__ ssq_t,
                        const float* __restrict__ diffp, const int* __restrict__ pnorm,
                        float* __restrict__ out) {
    if (threadIdx.x != 0 || blockIdx.x != 0) return;
    const int p = *pnorm;
    const float invp = 1.f / (float)p;
    const float fdim = (float)FDIM;
    auto rootp = [&](float x) { return (p == 2) ? sqrtf(x) : powf(x, invp); };

    // f_dist per domain
    float fd[2];
    for (int dom = 0; dom < 2; ++dom) {
        const int* cnt = dom ? cnt_t : cnt_s;
        const float* ssq = dom ? ssq_t : ssq_s;
        float acc = 0.f, nvalid = 0.f;
        for (int i = 0; i < NCLS; ++i) {
            bool valid = cnt[i] > 0;
            float ss = valid ? ssq[i] : 1.f;
            float per = rootp(ss) / (fmaxf((float)cnt[i], 1.f) * fdim);
            if (valid) { acc += per; nvalid += 1.f; }
        }
        fd[dom] = acc / nvalid;
    }

    // c_dist over seen classes
    float n_seen = 0.f;
    bool seen[NCLS];
    for (int i = 0; i < NCLS; ++i) {
        seen[i] = (cnt_s[i] + cnt_t[i]) > 0;
        if (seen[i]) n_seen += 1.f;
    }
    float cacc = 0.f;
    for (int i = 0; i < NCLS; ++i) {
        float row = 0.f;
        if (seen[i])
            for (int j = 0; j < NCLS; ++j)
                if (j != i && seen[j]) row += diffp[i * NCLS + j];
        float rs = seen[i] ? row : 1.f;
        float per = rootp(rs) / ((n_seen - 1.f) * fdim);
        if (seen[i]) cacc += per;
    }
    out[0] = cacc / n_seen;
    out[1] = fd[0];
    out[2] = fd[1];
}

extern "C" void kernel_launch(void* const* d_in, const int* in_sizes, int n_in,
                              void* d_out, int out_size, void* d_ws, size_t ws_size,
                              hipStream_t stream) {
    const float* src_feat = (const float*)d_in[0];
    const float* src_sm   = (const float*)d_in[1];
    const float* tgt_feat = (const float*)d_in[2];
    const float* tgt_sm   = (const float*)d_in[3];
    const int*   pnorm    = (const int*)d_in[4];
    float* out = (float*)d_out;

    float* w = (float*)d_ws;
    float* sum_s = w + OFF_SUM_S;
    float* sum_t = w + OFF_SUM_T;
    float* cent  = w + OFF_CENT;
    float* sq_s  = w + OFF_SQ_S;
    float* sq_t  = w + OFF_SQ_T;
    int*   cnt_s = (int*)(w + OFF_CNT_S);
    int*   cnt_t = (int*)(w + OFF_CNT_T);
    float* ssq_s = w + OFF_SSQ_S;
    float* ssq_t = w + OFF_SSQ_T;
    float* diffp = w + OFF_DIFFP;
    int*   lab_s = (int*)(w + OFF_LAB_S);
    int*   lab_t = (int*)(w + OFF_LAB_T);

    k_zero<<<(OFF_ZERO_END + 255) / 256, 256, 0, stream>>>(w, OFF_ZERO_END);

    k_labels<<<NPIX / 256, 256, 0, stream>>>(src_sm, lab_s, cnt_s);
    k_labels<<<NPIX / 256, 256, 0, stream>>>(tgt_sm, lab_t, cnt_t);

    dim3 gseg(FDIM / 64, PW);
    k_segsum<<<gseg, 256, 0, stream>>>(src_feat, lab_s, sum_s, sq_s);
    k_segsum<<<gseg, 256, 0, stream>>>(tgt_feat, lab_t, sum_t, sq_t);

    k_centroid<<<NCLS, 256, 0, stream>>>(sum_s, sum_t, cnt_s, cnt_t,
                                         sq_s, sq_t, cent, ssq_s, ssq_t, pnorm);

    k_pairwise<<<dim3(NCLS, NCLS), 256, 0, stream>>>(cent, diffp, pnorm);

    k_fdist_generic<<<NPIX / 256, 256, 0, stream>>>(src_feat, lab_s, cent, ssq_s, pnorm);
    k_fdist_generic<<<NPIX / 256, 256, 0, stream>>>(tgt_feat, lab_t, cent, ssq_t, pnorm);

    k_final<<<1, 1, 0, stream>>>(cnt_s, cnt_t, ssq_s, ssq_t, diffp, pnorm, out);
}